// GCNLayer_19902878450413
// MI455X (gfx1250) — compile-verified
//
#include <hip/hip_runtime.h>

typedef __attribute__((ext_vector_type(2))) float v2f;
typedef __attribute__((ext_vector_type(8))) float v8f;

#define GCN_DIN 128
#define GCN_DOUT 128

// ---------------- zero fill (graph-capture safe, no hipMemset) ----------------
__global__ void gcn_zero_f32(float* __restrict__ p, long long n) {
    long long i = (long long)blockIdx.x * blockDim.x + threadIdx.x;
    long long stride = (long long)gridDim.x * blockDim.x;
    for (; i < n; i += stride) p[i] = 0.0f;
}

// ---------------- degree: deg[row[e]] += 1 ----------------
__global__ void gcn_degree(const long long* __restrict__ row, float* __restrict__ deg, int E) {
    int e = blockIdx.x * blockDim.x + threadIdx.x;
    if (e < E) atomicAdd(&deg[(int)row[e]], 1.0f);
}

// ---------------- d = rsqrt(deg) ----------------
__global__ void gcn_rsqrt(const float* __restrict__ deg, float* __restrict__ d, int n) {
    int i = blockIdx.x * blockDim.x + threadIdx.x;
    if (i < n) d[i] = __frsqrt_rn(deg[i]);
}

// ---------------- h = x @ W^T + b via V_WMMA_F32_16X16X4_F32 ----------------
// One block = one 16-row strip. 8 waves per block, one 16-col tile each.
// K=128 processed as 32 WMMA steps of K=4.
// A-frag (16x4 f32, 2 VGPRs): lanes 0-15 -> M=lane,K={0,1}; lanes 16-31 -> M=lane-16,K={2,3}
// B-frag (4x16 f32, 2 VGPRs): mirrors C layout: VGPR0 holds rows K=0(K=2 in hi half), VGPR1 K=1(K=3)
// C/D (16x16 f32, 8 VGPRs): VGPR v: lanes 0-15 -> M=v, lanes 16-31 -> M=v+8; N=lane&15
__global__ void gcn_gemm_wmma(const float* __restrict__ x, const float* __restrict__ W,
                              const float* __restrict__ b, float* __restrict__ h, int N) {
    const int wave = threadIdx.x >> 5;
    const int lane = threadIdx.x & 31;
    const int half = lane >> 4;    // 0 or 1
    const int l    = lane & 15;
    const int rowBase = blockIdx.x * 16;
    if (rowBase + 16 > N) return;              // uniform per block; EXEC stays all-ones
    const int colBase = wave * 16;

    // A element stream: x[rowBase+l][4k + 2*half + {0,1}]
    const float* xp = x + (size_t)(rowBase + l) * GCN_DIN + 2 * half;
    // B element: B[k][n] = W^T[k][n] = W[n][k]; n = colBase+l, k = 4k + 2*half + {0,1}
    const float* wp = W + (size_t)(colBase + l) * GCN_DIN + 2 * half;

    v8f c = {};
#pragma unroll
    for (int k = 0; k < GCN_DIN / 4; ++k) {
        v2f a  = *(const v2f*)(xp + 4 * k);
        v2f bb = *(const v2f*)(wp + 4 * k);
        c = __builtin_amdgcn_wmma_f32_16x16x4_f32(
            /*neg_a=*/false, a, /*neg_b=*/false, bb,
            /*c_mod=*/(short)0, c, /*reuse_a=*/false, /*reuse_b=*/false);
    }

    const int n    = colBase + l;
    const float bv = b[n];
    const int mb   = rowBase + 8 * half;
#pragma unroll
    for (int v = 0; v < 8; ++v)
        h[(size_t)(mb + v) * GCN_DOUT + n] = c[v] + bv;
}

// ---------------- edge aggregation: out[row] += d[col] * h[col] ----------------
// One wave per edge (grid-stride). 128 floats/row = one float4 per lane.
// h and out are L2-resident (51 MB each vs 192 MB L2), so this is an L2
// gather + L2 atomic scatter.
__global__ void gcn_edge_agg(const long long* __restrict__ row, const long long* __restrict__ col,
                             const float* __restrict__ h, const float* __restrict__ d,
                             float* __restrict__ out, int E) {
    const int lane = threadIdx.x & 31;
    long long wid = (long long)blockIdx.x * (blockDim.x >> 5) + (threadIdx.x >> 5);
    long long nw  = (long long)gridDim.x * (blockDim.x >> 5);
    for (long long e = wid; e < E; e += nw) {
        int r = (int)row[e];
        int c = (int)col[e];
        float s = d[c];
        float4 v = ((const float4*)(h + (size_t)c * GCN_DOUT))[lane];
        float* o = out + (size_t)r * GCN_DOUT + lane * 4;
        atomicAdd(o + 0, s * v.x);
        atomicAdd(o + 1, s * v.y);
        atomicAdd(o + 2, s * v.z);
        atomicAdd(o + 3, s * v.w);
    }
}

// ---------------- out = d*out + d^2*h (in place, float4) ----------------
__global__ void gcn_finalize(float* __restrict__ out, const float* __restrict__ h,
                             const float* __restrict__ d, int N) {
    long long idx = (long long)blockIdx.x * blockDim.x + threadIdx.x;  // over N * 32 float4s
    long long total = (long long)N * (GCN_DOUT / 4);
    if (idx >= total) return;
    int i = (int)(idx >> 5);   // 32 float4 per row
    float dv = d[i];
    float dd = dv * dv;
    float4 o  = ((float4*)out)[idx];
    float4 hh = ((const float4*)h)[idx];
    o.x = dv * o.x + dd * hh.x;
    o.y = dv * o.y + dd * hh.y;
    o.z = dv * o.z + dd * hh.z;
    o.w = dv * o.w + dd * hh.w;
    ((float4*)out)[idx] = o;
}

extern "C" void kernel_launch(void* const* d_in, const int* in_sizes, int n_in,
                              void* d_out, int out_size, void* d_ws, size_t ws_size,
                              hipStream_t stream) {
    const float*     x  = (const float*)d_in[0];       // [N, 128] f32
    const long long* ei = (const long long*)d_in[1];   // [2, E] int64
    const float*     W  = (const float*)d_in[2];       // [128, 128] f32
    const float*     b  = (const float*)d_in[3];       // [128] f32
    float* out = (float*)d_out;                        // [N, 128] f32

    const int N = in_sizes[0] / GCN_DIN;               // 100000
    const int E = in_sizes[1] / 2;                     // 1600000
    const long long* rowIdx = ei;
    const long long* colIdx = ei + E;

    // workspace layout: h [N*128] | deg [N] | d [N]   (~52 MB)
    float* h   = (float*)d_ws;
    float* deg = h + (size_t)N * GCN_DOUT;
    float* dv  = deg + N;

    // 1) zero accumulators (out is the agg buffer; deg is atomic-accumulated)
    gcn_zero_f32<<<4096, 256, 0, stream>>>(out, (long long)N * GCN_DOUT);
    gcn_zero_f32<<<256, 256, 0, stream>>>(deg, (long long)N);

    // 2) h = x @ W^T + b   (fp32 WMMA; N divisible by 16 -> exact grid)
    gcn_gemm_wmma<<<(N + 15) / 16, 256, 0, stream>>>(x, W, b, h, N);

    // 3) degree + rsqrt
    gcn_degree<<<(E + 255) / 256, 256, 0, stream>>>(rowIdx, deg, E);
    gcn_rsqrt<<<(N + 255) / 256, 256, 0, stream>>>(deg, dv, N);

    // 4) edge aggregation into out (wave per edge, grid-stride; 65536 resident waves)
    gcn_edge_agg<<<8192, 256, 0, stream>>>(rowIdx, colIdx, h, dv, out, E);

    // 5) out = d*out + d^2*h
    long long nvec = (long long)N * (GCN_DOUT / 4);
    gcn_finalize<<<(int)((nvec + 255) / 256), 256, 0, stream>>>(out, h, dv, N);
}